// Voxelization_74345883893893
// MI455X (gfx1250) — compile-verified
//
#include <hip/hip_runtime.h>
#include <stdint.h>

typedef uint32_t u32;

// ---------------- problem constants (match reference) ----------------
constexpr int       B       = 2;
constexpr int       F       = 5;
constexpr int       GX      = 1440;
constexpr int       GY      = 1440;
constexpr int       GZ      = 40;
constexpr u32       GXY     = (u32)GX * (u32)GY;        // 2,073,600
constexpr long long NVOX    = (long long)GXY * GZ;      // 82,944,000
constexpr int       NW      = (int)(NVOX / 32);         // 2,592,000 (exact)
constexpr int       MAX_VOX = 120000;
constexpr int       MAX_PTS = 10;
constexpr int       WPB     = 1024;                     // bitmap words per scan block
constexpr int       NBLK    = (NW + WPB - 1) / WPB;     // 2532
constexpr int       NBLKA   = (NBLK + 3) & ~3;          // 2532 (16B aligned)
constexpr u32       BAD_ID  = 0xFFFFFFFFu;
constexpr int       IMAXV   = 0x7FFFFFFF;

// output layout (flat floats, reference tuple order)
constexpr size_t VOX_ELEMS   = (size_t)B * MAX_VOX * MAX_PTS * F;  // 12,000,000
constexpr size_t COORD_ELEMS = (size_t)B * MAX_VOX * 4;            //    960,000
constexpr size_t NPTS_OFF    = VOX_ELEMS + COORD_ELEMS;
constexpr size_t OUT_TOTAL   = NPTS_OFF + (size_t)B * MAX_VOX;     // 13,200,000

// ---------------- small utility kernels ----------------
__global__ void k_fill_u32(u32* __restrict__ p, size_t n, u32 v) {
  for (size_t i = (size_t)blockIdx.x * blockDim.x + threadIdx.x; i < n;
       i += (size_t)gridDim.x * blockDim.x)
    p[i] = v;
}

__global__ void k_init_out(float* __restrict__ out) {
  for (size_t i = (size_t)blockIdx.x * blockDim.x + threadIdx.x; i < OUT_TOTAL;
       i += (size_t)gridDim.x * blockDim.x) {
    float v = 0.0f;
    if (i >= VOX_ELEMS && i < NPTS_OFF) {          // coords region: [b,-1,-1,-1]
      size_t j = i - VOX_ELEMS;
      int comp = (int)(j & 3);
      int row  = (int)(j >> 2);
      v = (comp == 0) ? (float)(row / MAX_VOX) : -1.0f;
    }
    out[i] = v;
  }
}

// async global->LDS staging of a block of point records (CDNA5 ASYNC path)
__device__ __forceinline__ void stage_points_async(const float* gbase, float* sp,
                                                   int cnt, int t) {
  unsigned lds0 = (unsigned)(uintptr_t)sp;         // low 32 bits = LDS byte offset
  int totaldw = cnt * F;
  int nseg = totaldw >> 2;                          // 16B segments
  int rem  = totaldw & 3;
  for (int s = t; s < nseg; s += 256) {
    unsigned laddr = lds0 + (unsigned)(s * 16);
    const float* g = gbase + (size_t)s * 4;
    asm volatile("global_load_async_to_lds_b128 %0, %1, off"
                 :: "v"(laddr), "v"(g) : "memory");
  }
  if (t < rem) {
    unsigned laddr = lds0 + (unsigned)(nseg * 16 + t * 4);
    const float* g = gbase + (size_t)nseg * 4 + t;
    asm volatile("global_load_async_to_lds_b32 %0, %1, off"
                 :: "v"(laddr), "v"(g) : "memory");
  }
  asm volatile("s_wait_asynccnt 0x0" ::: "memory");
  __syncthreads();
}

// ---------------- pass 1: mark occupancy bitmap, cache ids ----------------
__global__ void k_mark(const float* __restrict__ pts, u32* __restrict__ bitmap,
                       u32* __restrict__ ids, int N) {
  __shared__ float sp[256 * F];
  int b    = blockIdx.y;
  int base = blockIdx.x * 256;
  int cnt  = min(256, N - base);
  int t    = threadIdx.x;
  const float* gbase = pts + ((size_t)b * N + base) * F;
  stage_points_async(gbase, sp, cnt, t);
  if (t >= cnt) return;

  float x = sp[t * F + 0], y = sp[t * F + 1], z = sp[t * F + 2];
  int cx = (int)floorf((x - (-54.0f)) / 0.075f);
  int cy = (int)floorf((y - (-54.0f)) / 0.075f);
  int cz = (int)floorf((z - (-5.0f))  / 0.2f);
  u32 id = BAD_ID;
  if ((unsigned)cx < (unsigned)GX && (unsigned)cy < (unsigned)GY &&
      (unsigned)cz < (unsigned)GZ) {
    id = (u32)cz * GXY + (u32)cy * (u32)GX + (u32)cx;
    atomicOr(&bitmap[(size_t)b * NW + (id >> 5)], 1u << (id & 31u));
  }
  ids[(size_t)b * N + base + t] = id;
}

// ---------------- pass 2a: per-block popcount sums ----------------
__global__ void k_bsum(const u32* __restrict__ bitmap, u32* __restrict__ bsum) {
  __shared__ u32 red[256];
  int b = blockIdx.y;
  int t = threadIdx.x;
  const u32* bm = bitmap + (size_t)b * NW;
  size_t w = (size_t)blockIdx.x * WPB + (size_t)t * 4;
  u32 s = 0;
  if (w + 3 < (size_t)NW) {
    uint4 v = *(const uint4*)(bm + w);
    s = __popc(v.x) + __popc(v.y) + __popc(v.z) + __popc(v.w);
  } else {
    for (int k = 0; k < 4; ++k)
      if (w + k < (size_t)NW) s += __popc(bm[w + k]);
  }
  red[t] = s;
  __syncthreads();
  for (int off = 128; off > 0; off >>= 1) {
    if (t < off) red[t] += red[t + off];
    __syncthreads();
  }
  if (t == 0) bsum[(size_t)b * NBLKA + blockIdx.x] = red[0];
}

// ---------------- pass 2b: exclusive scan of block sums (1 block/batch) ----
__global__ void k_scan(u32* __restrict__ bsum) {
  __shared__ u32 s[256];
  int b = blockIdx.x;
  u32* p = bsum + (size_t)b * NBLKA;
  int t = threadIdx.x;
  u32 carry = 0;
  for (int base = 0; base < NBLK; base += 256) {
    u32 v = (base + t < NBLK) ? p[base + t] : 0u;
    __syncthreads();
    s[t] = v;
    __syncthreads();
    for (int off = 1; off < 256; off <<= 1) {
      u32 u = (t >= off) ? s[t - off] : 0u;
      __syncthreads();
      s[t] += u;
      __syncthreads();
    }
    u32 incl = s[t];
    u32 tot  = s[255];
    if (base + t < NBLK) p[base + t] = carry + incl - v;   // exclusive
    carry += tot;
  }
}

// ---------------- pass 2c: per-word exclusive rank prefix ----------------
__global__ void k_wpref(const u32* __restrict__ bitmap, const u32* __restrict__ bsum,
                        u32* __restrict__ wpref) {
  __shared__ u32 sc[256];
  int b = blockIdx.y;
  int t = threadIdx.x;
  const u32* bm = bitmap + (size_t)b * NW;
  u32* wp = wpref + (size_t)b * NW;
  size_t w = (size_t)blockIdx.x * WPB + (size_t)t * 4;
  u32 c0 = 0, c1 = 0, c2 = 0, c3 = 0;
  if (w + 3 < (size_t)NW) {
    uint4 v = *(const uint4*)(bm + w);
    c0 = __popc(v.x); c1 = __popc(v.y); c2 = __popc(v.z); c3 = __popc(v.w);
  } else {
    if (w + 0 < (size_t)NW) c0 = __popc(bm[w + 0]);
    if (w + 1 < (size_t)NW) c1 = __popc(bm[w + 1]);
    if (w + 2 < (size_t)NW) c2 = __popc(bm[w + 2]);
    if (w + 3 < (size_t)NW) c3 = __popc(bm[w + 3]);
  }
  u32 mysum = c0 + c1 + c2 + c3;
  sc[t] = mysum;
  __syncthreads();
  for (int off = 1; off < 256; off <<= 1) {
    u32 u = (t >= off) ? sc[t - off] : 0u;
    __syncthreads();
    sc[t] += u;
    __syncthreads();
  }
  u32 run = sc[t] - mysum + bsum[(size_t)b * NBLKA + blockIdx.x];  // exclusive
  if (w + 0 < (size_t)NW) { wp[w + 0] = run; run += c0; }
  if (w + 1 < (size_t)NW) { wp[w + 1] = run; run += c1; }
  if (w + 2 < (size_t)NW) { wp[w + 2] = run; run += c2; }
  if (w + 3 < (size_t)NW) { wp[w + 3] = run; }
}

// ---------------- pass 3: write coords for rank < MAX_VOX ----------------
__global__ void k_coords(const u32* __restrict__ bitmap, const u32* __restrict__ wpref,
                         float* __restrict__ out) {
  int b = blockIdx.y;
  size_t w = (size_t)blockIdx.x * 256 + threadIdx.x;
  if (w >= (size_t)NW) return;
  u32 bits = bitmap[(size_t)b * NW + w];
  if (!bits) return;
  u32 rank = wpref[(size_t)b * NW + w];
  float* coords = out + VOX_ELEMS;
  while (bits) {
    int bit = __ffs(bits) - 1;
    bits &= bits - 1;
    if (rank >= (u32)MAX_VOX) break;        // ranks ascend within the word
    u32 id  = (u32)w * 32u + (u32)bit;
    u32 zc  = id / GXY;
    u32 rem = id - zc * GXY;
    u32 yc  = rem / (u32)GX;
    u32 xc  = rem - yc * (u32)GX;
    size_t row = ((size_t)b * MAX_VOX + rank) * 4;
    coords[row + 0] = (float)b;
    coords[row + 1] = (float)zc;
    coords[row + 2] = (float)yc;
    coords[row + 3] = (float)xc;
    ++rank;
  }
}

// ------- pass 4: deterministic slot assignment (atomicMin cascade) -------
__global__ void k_insert(const u32* __restrict__ ids, const u32* __restrict__ bitmap,
                         const u32* __restrict__ wpref, int* __restrict__ slots, int N) {
  int b = blockIdx.y;
  int i = blockIdx.x * 256 + threadIdx.x;
  if (i >= N) return;
  u32 id = ids[(size_t)b * N + i];
  if (id == BAD_ID) return;
  u32 w = id >> 5, bit = id & 31u;
  u32 rank = wpref[(size_t)b * NW + w] +
             (u32)__popc(bitmap[(size_t)b * NW + w] & ((1u << bit) - 1u));
  if (rank >= (u32)MAX_VOX) return;
  int* sl = slots + ((size_t)b * MAX_VOX + rank) * MAX_PTS;
  __builtin_prefetch(sl, 1, 3);               // global_prefetch_b8 to warm L2 line
  int x = i;                                   // keep the MAX_PTS smallest indices,
  for (int s = 0; s < MAX_PTS; ++s) {          // sorted ascending == reference's
    int old = atomicMin(&sl[s], x);            // stable within-voxel slot order
    if (old == IMAXV) break;
    x = (old > x) ? old : x;
  }
}

// ---------------- pass 5: gather features + npts ----------------
__global__ void k_gather(const float* __restrict__ pts, const int* __restrict__ slots,
                         float* __restrict__ out, int N) {
  int b = blockIdx.y;
  int v = blockIdx.x * 256 + threadIdx.x;
  if (v >= MAX_VOX) return;
  const int* sl = slots + ((size_t)b * MAX_VOX + v) * MAX_PTS;
  float* vox = out + ((size_t)b * MAX_VOX + v) * (size_t)(MAX_PTS * F);
  int n = 0;
  for (int s = 0; s < MAX_PTS; ++s) {
    int idx = sl[s];
    if (idx == IMAXV) break;
    ++n;
    const float* src = pts + ((size_t)b * N + idx) * F;
#pragma unroll
    for (int f = 0; f < F; ++f) vox[s * F + f] = src[f];
  }
  out[NPTS_OFF + (size_t)b * MAX_VOX + v] = (float)n;
}

// ---------------- launcher ----------------
extern "C" void kernel_launch(void* const* d_in, const int* in_sizes, int n_in,
                              void* d_out, int out_size, void* d_ws, size_t ws_size,
                              hipStream_t stream) {
  const float* pts = (const float*)d_in[0];
  int N = in_sizes[0] / (B * F);               // 200000
  float* out = (float*)d_out;

  u32* bitmap = (u32*)d_ws;                    // [B][NW]
  u32* wpref  = bitmap + (size_t)B * NW;       // [B][NW]
  u32* bsum   = wpref  + (size_t)B * NW;       // [B][NBLKA]
  u32* ids    = bsum   + (size_t)B * NBLKA;    // [B][N]
  int* slots  = (int*)(ids + (size_t)B * N);   // [B][MAX_VOX][MAX_PTS]

  size_t need = ((size_t)B * NW * 2 + (size_t)B * NBLKA + (size_t)B * N +
                 (size_t)B * MAX_VOX * MAX_PTS) * sizeof(u32);
  if (ws_size < need || (size_t)out_size < OUT_TOTAL || N <= 0) return;

  int pblocks = (N + 255) / 256;

  k_fill_u32<<<2048, 256, 0, stream>>>(bitmap, (size_t)B * NW, 0u);
  k_fill_u32<<<64,   256, 0, stream>>>(bsum, (size_t)B * NBLKA, 0u);
  k_fill_u32<<<2048, 256, 0, stream>>>((u32*)slots,
                                       (size_t)B * MAX_VOX * MAX_PTS, (u32)IMAXV);
  k_init_out<<<4096, 256, 0, stream>>>(out);

  k_mark  <<<dim3(pblocks, B), 256, 0, stream>>>(pts, bitmap, ids, N);
  k_bsum  <<<dim3(NBLK, B),    256, 0, stream>>>(bitmap, bsum);
  k_scan  <<<B,                256, 0, stream>>>(bsum);
  k_wpref <<<dim3(NBLK, B),    256, 0, stream>>>(bitmap, bsum, wpref);
  k_coords<<<dim3((NW + 255) / 256, B), 256, 0, stream>>>(bitmap, wpref, out);
  k_insert<<<dim3(pblocks, B), 256, 0, stream>>>(ids, bitmap, wpref, slots, N);
  k_gather<<<dim3((MAX_VOX + 255) / 256, B), 256, 0, stream>>>(pts, slots, out, N);
}